// PoseOptLayer_4861902979104
// MI455X (gfx1250) — compile-verified
//
#include <hip/hip_runtime.h>
#include <cstdint>

#define NJ  24
#define TPB 128

typedef __attribute__((ext_vector_type(2))) float v2f;
typedef __attribute__((ext_vector_type(8))) float v8f;

// Builtin expects: (int __vector(4) addrspace(1)*, <lds ptr>, imm offset, imm cpol)
typedef __attribute__((__vector_size__(16))) int v4i_t;
typedef __attribute__((address_space(1))) v4i_t* as1_v4i;
typedef __attribute__((address_space(3))) v4i_t* as3_v4i;

// ---------------- CDNA5 async global->LDS copy (16B), tracked on ASYNCcnt ----
__device__ __forceinline__ void async_copy_b128(const float* g, float* l) {
#if __has_builtin(__builtin_amdgcn_global_load_async_to_lds_b128)
  // uintptr_t route: global generic ptr value == AS1 address; low 32 bits of a
  // generic LDS ptr == LDS offset (addrspacecast 3->0 is aperture|offset).
  __builtin_amdgcn_global_load_async_to_lds_b128(
      (as1_v4i)(uintptr_t)g, (as3_v4i)(uintptr_t)l, 0, 0);
#else
  unsigned lo = (unsigned)(uintptr_t)l; // low 32 bits = LDS offset
  asm volatile("global_load_async_to_lds_b128 %0, %1, off"
               :: "v"(lo), "v"(g) : "memory");
#endif
}

__device__ __forceinline__ void wait_asynccnt0() {
#if __has_builtin(__builtin_amdgcn_s_wait_asynccnt)
  __builtin_amdgcn_s_wait_asynccnt(0);
#else
  asm volatile("s_wait_asynccnt 0x0" ::: "memory");
#endif
}

// parent[j] for j>=1 (== JOINT_TREES[j]); parent[0] unused
#define PAR_INIT {0,0,0,0,1,2,3,4,5,6,7,8,9,9,9,12,13,14,16,17,18,19,20,21}

__global__ __launch_bounds__(TPB) void pose_chain_kernel(
    const float* __restrict__ pelvis, const float* __restrict__ bones,
    const float* __restrict__ rest,   const int* __restrict__ idxs,
    float* __restrict__ out_kp,   float* __restrict__ out_bone,
    float* __restrict__ out_skts, float* __restrict__ out_l2w,
    float* __restrict__ out_rot)
{
  __shared__ int   s_idx[TPB];
  __shared__ __align__(16) float s_rest[NJ * 3];
  __shared__ __align__(16) float s_bones[TPB * NJ * 3];
  // per-element 24 slots of 12 floats [R row0|t0, R row1|t1, R row2|t2];
  // holds rel first, overwritten in place by l2w as the chain is composed.
  __shared__ __align__(16) float s_slot[TPB * NJ * 12 + 16];

  const int  t = threadIdx.x;
  const long e = (long)blockIdx.x * TPB + t;
  const int  PAR[NJ] = PAR_INIT;

  // ---------------- Phase 0: stage idx, rest_pose; async-gather bones -------
  s_idx[t] = idxs[e];
  if (t < NJ * 3) s_rest[t] = rest[t];
  __syncthreads();

  // 18 x 16B chunks per element; consecutive lanes fetch consecutive chunks
  #pragma unroll
  for (int i = 0; i < 18; ++i) {
    int k  = t + i * TPB;            // 0 .. TPB*18-1
    int el = k / 18, c = k % 18;
    async_copy_b128(bones + (long)s_idx[el] * 72 + c * 4,
                    &s_bones[el * 72 + c * 4]);
  }
  wait_asynccnt0();
  __syncthreads();

  const int   ix = s_idx[t];
  const float px = pelvis[ix * 3 + 0];
  const float py = pelvis[ix * 3 + 1];
  const float pz = pelvis[ix * 3 + 2];

  // ---------------- Phase 1: bone passthrough, Rodrigues, rel into LDS ------
  {
    float* ob = out_bone + e * 72;
    const float* sb = &s_bones[t * 72];
    #pragma unroll
    for (int i = 0; i < 18; ++i)
      ((float4*)ob)[i] = ((const float4*)sb)[i];
  }

  #pragma unroll
  for (int j = 0; j < NJ; ++j) {
    float vx = s_bones[t * 72 + j * 3 + 0];
    float vy = s_bones[t * 72 + j * 3 + 1];
    float vz = s_bones[t * 72 + j * 3 + 2];
    float w  = vx * vx + vy * vy + vz * vz + 1e-8f;
    float ir = rsqrtf(w);
    float th = w * ir;                      // theta = sqrt(w)
    float s  = __sinf(th), c = __cosf(th), oc = 1.0f - c;
    float ax = vx * ir, ay = vy * ir, az = vz * ir;
    // R = I + s*K + (1-c)*K^2,  K = skew(axis)
    float R00 = 1.0f - oc * (ay * ay + az * az);
    float R01 = -s * az + oc * ax * ay;
    float R02 =  s * ay + oc * ax * az;
    float R10 =  s * az + oc * ax * ay;
    float R11 = 1.0f - oc * (ax * ax + az * az);
    float R12 = -s * ax + oc * ay * az;
    float R20 = -s * ay + oc * ax * az;
    float R21 =  s * ax + oc * ay * az;
    float R22 = 1.0f - oc * (ax * ax + ay * ay);

    { // rots output
      float* orr = out_rot + ((long)e * NJ + j) * 9;
      orr[0]=R00; orr[1]=R01; orr[2]=R02;
      orr[3]=R10; orr[4]=R11; orr[5]=R12;
      orr[6]=R20; orr[7]=R21; orr[8]=R22;
    }

    float tx, ty, tz;
    if (j == 0) { tx = s_rest[0]; ty = s_rest[1]; tz = s_rest[2]; }
    else {
      int p = PAR[j];
      tx = s_rest[j*3+0] - s_rest[p*3+0];
      ty = s_rest[j*3+1] - s_rest[p*3+1];
      tz = s_rest[j*3+2] - s_rest[p*3+2];
    }
    const int sb = (t * NJ + j) * 12;
    s_slot[sb+0]=R00; s_slot[sb+1]=R01; s_slot[sb+ 2]=R02; s_slot[sb+ 3]=tx;
    s_slot[sb+4]=R10; s_slot[sb+5]=R11; s_slot[sb+ 6]=R12; s_slot[sb+ 7]=ty;
    s_slot[sb+8]=R20; s_slot[sb+9]=R21; s_slot[sb+10]=R22; s_slot[sb+11]=tz;
  }
  __syncthreads();

  // ---------------- Phase 2: chain compose l2w[j] = l2w[par] * rel[j] -------
#if __has_builtin(__builtin_amdgcn_wmma_f32_16x16x4_f32)
  {
    // Wave-cooperative: 4 elements per WMMA (block-diagonal of 16x16 = 4x (4x4)).
    const int lane = t & 31;
    const int wv   = t >> 5;
    const int half = lane >> 4;   // K-half: vgpr0 holds K=2*half, vgpr1 K=2*half+1
    const int q    = lane & 15;
    const int m    = q >> 2;      // 4x4 block index (element within group)
    const int rc   = q & 3;       // row (A) / col (B,D) within block
    #pragma unroll 1
    for (int g = 0; g < 8; ++g) {
      const int elm = wv * 32 + g * 4 + m;
      #pragma unroll
      for (int j = 1; j < NJ; ++j) {
        const int p     = PAR[j];
        const int pbase = (elm * NJ + p) * 12;
        const int jbase = (elm * NJ + j) * 12;
        const int k0    = half * 2;
        // A(16x4): rows 4m+rc = parent l2w rows; bottom row = (0,0,0,1)
        float a0, a1;
        if (rc < 3) {
          a0 = s_slot[pbase + rc * 4 + k0];
          a1 = s_slot[pbase + rc * 4 + k0 + 1];
        } else {
          a0 = 0.0f;
          a1 = (half == 1) ? 1.0f : 0.0f;   // K==3 -> 1
        }
        // B(4x16): cols 4m+rc = rel cols; row K==3 -> (0,0,0,1)
        float b0 = s_slot[jbase + k0 * 4 + rc];                     // K = 0 or 2
        float b1 = (half == 0) ? s_slot[jbase + 4 + rc]             // K = 1
                               : ((rc == 3) ? 1.0f : 0.0f);         // K = 3
        v2f A;  A.x  = a0; A.y  = a1;
        v2f Bv; Bv.x = b0; Bv.y = b1;
        v8f C = {0.f,0.f,0.f,0.f,0.f,0.f,0.f,0.f};
        v8f D = __builtin_amdgcn_wmma_f32_16x16x4_f32(false, A, false, Bv,
                                                      (short)0, C, false, false);
        // diag block (m,m): rows 4m..4m+2 live in vgprs (m&1)*4.. when (m>>1)==half
        if ((m >> 1) == half) {
          float r0 = (m & 1) ? D[4] : D[0];
          float r1 = (m & 1) ? D[5] : D[1];
          float r2 = (m & 1) ? D[6] : D[2];
          s_slot[jbase + 0 + rc] = r0;
          s_slot[jbase + 4 + rc] = r1;
          s_slot[jbase + 8 + rc] = r2;
        }
        __builtin_amdgcn_wave_barrier();   // keep LDS RAW order across lanes
      }
    }
  }
#else
  { // VALU fallback: per-thread sequential rigid compose
    float* sl = &s_slot[t * NJ * 12];
    #pragma unroll
    for (int j = 1; j < NJ; ++j) {
      const float* P = sl + PAR[j] * 12;
      float*       R = sl + j * 12;
      float o[12];
      #pragma unroll
      for (int r = 0; r < 3; ++r) {
        #pragma unroll
        for (int c = 0; c < 4; ++c) {
          float acc = P[r*4+0]*R[0+c] + P[r*4+1]*R[4+c] + P[r*4+2]*R[8+c];
          if (c == 3) acc += P[r*4+3];
          o[r*4+c] = acc;
        }
      }
      #pragma unroll
      for (int i = 0; i < 12; ++i) R[i] = o[i];
    }
  }
#endif
  __syncthreads();

  // ---------------- Phase 3: add pelvis, analytic rigid inverse, store ------
  {
    const float* sl  = &s_slot[t * NJ * 12];
    float* okp = out_kp  + e * 72;
    float* ol2 = out_l2w + e * 384;
    float* osk = out_skts + e * 384;
    #pragma unroll
    for (int j = 0; j < NJ; ++j) {
      float4 q0 = *(const float4*)(sl + j * 12 + 0);
      float4 q1 = *(const float4*)(sl + j * 12 + 4);
      float4 q2 = *(const float4*)(sl + j * 12 + 8);
      float tx = q0.w + px, ty = q1.w + py, tz = q2.w + pz;
      // l2ws (row major 4x4)
      float4* L = (float4*)(ol2 + j * 16);
      L[0] = make_float4(q0.x, q0.y, q0.z, tx);
      L[1] = make_float4(q1.x, q1.y, q1.z, ty);
      L[2] = make_float4(q2.x, q2.y, q2.z, tz);
      L[3] = make_float4(0.f, 0.f, 0.f, 1.f);
      // skts = inv(l2w) = [R^T | -R^T t]
      float m0 = -(q0.x * tx + q1.x * ty + q2.x * tz);
      float m1 = -(q0.y * tx + q1.y * ty + q2.y * tz);
      float m2 = -(q0.z * tx + q1.z * ty + q2.z * tz);
      float4* S = (float4*)(osk + j * 16);
      S[0] = make_float4(q0.x, q1.x, q2.x, m0);
      S[1] = make_float4(q0.y, q1.y, q2.y, m1);
      S[2] = make_float4(q0.z, q1.z, q2.z, m2);
      S[3] = make_float4(0.f, 0.f, 0.f, 1.f);
      okp[j * 3 + 0] = tx; okp[j * 3 + 1] = ty; okp[j * 3 + 2] = tz;
    }
  }
}

extern "C" void kernel_launch(void* const* d_in, const int* in_sizes, int n_in,
                              void* d_out, int out_size, void* d_ws, size_t ws_size,
                              hipStream_t stream) {
  (void)n_in; (void)out_size; (void)d_ws; (void)ws_size;
  const float* pelvis = (const float*)d_in[0];
  const float* bones  = (const float*)d_in[1];
  const float* rest   = (const float*)d_in[2];
  const int*   idxs   = (const int*)d_in[3];
  const long   Bn     = in_sizes[3];          // 131072, divisible by TPB

  float* out    = (float*)d_out;              // concat: kp,bone,skts,l2ws,rots
  float* o_kp   = out;
  float* o_bone = o_kp   + Bn * 72;
  float* o_skts = o_bone + Bn * 72;
  float* o_l2w  = o_skts + Bn * 384;
  float* o_rot  = o_l2w  + Bn * 384;

  dim3 grid((unsigned)((Bn + TPB - 1) / TPB)), block(TPB);
  hipLaunchKernelGGL(pose_chain_kernel, grid, block, 0, stream,
                     pelvis, bones, rest, idxs,
                     o_kp, o_bone, o_skts, o_l2w, o_rot);
}